// ModelTorch_28681791602778
// MI455X (gfx1250) — compile-verified
//
#include <hip/hip_runtime.h>

// out[pid] = verified_id[nd*pid + accept_lens[pid] - 1]
//
// Bandwidth-bound gather (~144 MB effective -> ~6 us @ 23.3 TB/s).
// Hot kernel is branch-free, all per-lane address math in u32 so loads/stores
// use the SGPR-base + u32-VGPR-offset (GVS) form:
//   1x global_load_b128 (4 lens), 4x clause-batched global_load_b32 gathers,
//   1x global_store_b128. Tail handled by a separate guarded kernel that only
// launches when bs % 1024 != 0 (never for the reference shape).

__device__ __forceinline__ const void* byte_off(const void* p, unsigned b) {
    // base_ptr + zext(u32) -> saddr + voffset addressing
    return (const void*)((const char*)p + (size_t)b);
}

__global__ __launch_bounds__(256) void gather_vec4_kernel(
    const float* __restrict__ verified_id,
    const int*   __restrict__ accept_lens,
    const int*   __restrict__ num_draft_p,   // 1-element device scalar
    float*       __restrict__ out)
{
    const unsigned nd   = (unsigned)num_draft_p[0];        // scalar broadcast load
    const unsigned t    = blockIdx.x * blockDim.x + threadIdx.x;
    const unsigned pid0 = t << 2;                          // 4 pids per thread
    const unsigned pb   = pid0 << 2;                       // byte offset for lens/out

    // One coalesced 16B load of 4 accepted lengths (saddr + u32 voffset).
    const int4 lens = *(const int4*)byte_off(accept_lens, pb);

    // All offsets fit in u32: max is 2^21 * 16 * 4 B = 128 MB.
    const unsigned base = pid0 * nd;

    const unsigned o0 = (base                    + (unsigned)(lens.x - 1)) << 2;
    const unsigned o1 = (base + nd               + (unsigned)(lens.y - 1)) << 2;
    const unsigned o2 = (base + 2u * nd          + (unsigned)(lens.z - 1)) << 2;
    const unsigned o3 = (base + 3u * nd          + (unsigned)(lens.w - 1)) << 2;

    // 4 independent gathers — clause-batched, single s_wait_loadcnt.
    const float r0 = *(const float*)byte_off(verified_id, o0);
    const float r1 = *(const float*)byte_off(verified_id, o1);
    const float r2 = *(const float*)byte_off(verified_id, o2);
    const float r3 = *(const float*)byte_off(verified_id, o3);

    float4 r;
    r.x = r0; r.y = r1; r.z = r2; r.w = r3;
    *(float4*)byte_off(out, pb) = r;                       // coalesced 16B store
}

__global__ __launch_bounds__(256) void gather_tail_kernel(
    const float* __restrict__ verified_id,
    const int*   __restrict__ accept_lens,
    const int*   __restrict__ num_draft_p,
    float*       __restrict__ out,
    int start, int bs)
{
    const int pid = start + blockIdx.x * blockDim.x + threadIdx.x;
    if (pid < bs) {
        const int nd  = num_draft_p[0];
        const int len = accept_lens[pid];
        out[pid] = verified_id[(long long)pid * nd + (len - 1)];
    }
}

extern "C" void kernel_launch(void* const* d_in, const int* in_sizes, int n_in,
                              void* d_out, int out_size, void* d_ws, size_t ws_size,
                              hipStream_t stream) {
    const float* verified_id = (const float*)d_in[0];
    const int*   accept_lens = (const int*)d_in[1];
    const int*   num_draft_p = (const int*)d_in[2];
    float*       out         = (float*)d_out;

    const int bs = in_sizes[1];              // accept_lens count == output count

    constexpr int kThreads     = 256;        // 8 wave32 per block
    constexpr int kElemsPerBlk = kThreads * 4;

    const int full_blocks = bs / kElemsPerBlk;        // branch-free coverage
    const int covered     = full_blocks * kElemsPerBlk;

    if (full_blocks > 0) {
        gather_vec4_kernel<<<full_blocks, kThreads, 0, stream>>>(
            verified_id, accept_lens, num_draft_p, out);
    }

    const int rem = bs - covered;                     // 0 for bs = 2^21
    if (rem > 0) {
        const int tail_blocks = (rem + kThreads - 1) / kThreads;
        gather_tail_kernel<<<tail_blocks, kThreads, 0, stream>>>(
            verified_id, accept_lens, num_draft_p, out, covered, bs);
    }
}